// LenChanger_14894946583337
// MI455X (gfx1250) — compile-verified
//
#include <hip/hip_runtime.h>
#include <stdint.h>

// ---------------------------------------------------------------------------
// LenChanger (duration-based frame expansion) for gfx1250 / MI455X.
// Pure data-movement op: ~145 MB of traffic, ~6 us at 23.3 TB/s. No WMMA
// applies (0 FLOPs); the CDNA5-specific paths used here are async
// global->LDS staging (ASYNCcnt) and non-temporal b128 streaming stores.
// ---------------------------------------------------------------------------

typedef float  vfloat4 __attribute__((ext_vector_type(4)));
typedef int    vint4   __attribute__((ext_vector_type(4)));
typedef int    gvint4  __attribute__((vector_size(16)));  // matches builtin param

#define AS1 __attribute__((address_space(1)))
#define AS3 __attribute__((address_space(3)))

#if __has_builtin(__builtin_amdgcn_global_load_async_to_lds_b128)
#define HAVE_ASYNC_LDS 1
#else
#define HAVE_ASYNC_LDS 0
#endif

static __device__ __forceinline__ void wait_asynccnt0() {
#if __has_builtin(__builtin_amdgcn_s_wait_asynccnt)
  __builtin_amdgcn_s_wait_asynccnt(0);
#else
  asm volatile("s_wait_asynccnt 0" ::: "memory");
#endif
}

static constexpr int kB = 16;
static constexpr int kT = 1024;
static constexpr int kD = 512;

// ---------------------------------------------------------------------------
// Kernel 1: per-row inclusive cumsum of durations.
// One 256-thread block per batch row; 4 elements/thread + Hillis-Steele scan.
// ---------------------------------------------------------------------------
__global__ __launch_bounds__(256) void lenchg_scan_kernel(
    const int* __restrict__ ds, int* __restrict__ csum) {
  const int b   = blockIdx.x;
  const int tid = threadIdx.x;  // 0..255, each owns 4 consecutive t's

  __shared__ int part[256];

  const int* row = ds + (size_t)b * kT;
  vint4 d = ((const vint4*)row)[tid];

  // local inclusive
  int s0 = d.x;
  int s1 = s0 + d.y;
  int s2 = s1 + d.z;
  int s3 = s2 + d.w;

  part[tid] = s3;
  __syncthreads();

  // Hillis-Steele inclusive scan over the 256 thread totals
  for (int off = 1; off < 256; off <<= 1) {
    int add = (tid >= off) ? part[tid - off] : 0;
    __syncthreads();
    part[tid] += add;
    __syncthreads();
  }

  const int base = part[tid] - s3;  // exclusive prefix for this thread's chunk

  vint4 c;
  c.x = base + s0;
  c.y = base + s1;
  c.z = base + s2;
  c.w = base + s3;
  ((vint4*)(csum + (size_t)b * kT))[tid] = c;
}

// ---------------------------------------------------------------------------
// Kernel 2: expansion. grid = (ceil(t_out/16), B), block = 256 threads.
// Stage the row's 4 KB cumsum into LDS (async global->LDS when available),
// then 128 threads per output frame: binary-search the source frame index
// (uniform within the frame's waves -> LDS broadcast reads), copy 2 KB with
// float4 loads and non-temporal float4 stores; zero-pad past totals[b].
// ---------------------------------------------------------------------------
#define FRAMES_PER_BLOCK 16

__global__ __launch_bounds__(256) void lenchg_expand_kernel(
    const float* __restrict__ xs, const int* __restrict__ csum,
    float* __restrict__ out, int t_out) {
  const int b   = blockIdx.y;
  const int tid = threadIdx.x;

  __shared__ int s_csum[kT];  // 4 KB

  const int* grow = csum + (size_t)b * kT;

  // --- Stage cumsum row into LDS: 256 threads x 16B = 4 KB ------------------
#if HAVE_ASYNC_LDS
  {
    AS1 gvint4* gsrc = (AS1 gvint4*)(grow + tid * 4);
    AS3 gvint4* ldst = (AS3 gvint4*)(&s_csum[tid * 4]);
    __builtin_amdgcn_global_load_async_to_lds_b128(gsrc, ldst, 0, 0);
    wait_asynccnt0();  // this wave's async loads have landed in LDS
  }
#else
  ((vint4*)s_csum)[tid] = ((const vint4*)grow)[tid];
#endif
  __syncthreads();

  const int total = s_csum[kT - 1];

  const int sub   = tid >> 7;    // which of 2 concurrent frames
  const int lane  = tid & 127;   // float4 slot within the 512-float frame
  const int fbase = blockIdx.x * FRAMES_PER_BLOCK;

  const float* xrow = xs + (size_t)b * kT * kD;
  float*       orow = out + (size_t)b * t_out * kD;

#pragma unroll
  for (int i = 0; i < FRAMES_PER_BLOCK; i += 2) {
    const int p = fbase + i + sub;
    if (p < t_out) {
      vfloat4 v = {0.0f, 0.0f, 0.0f, 0.0f};
      if (p < total) {
        // upper_bound: first idx with csum[idx] > p  (searchsorted 'right')
        int lo = 0, hi = kT;
        while (lo < hi) {
          int mid = (lo + hi) >> 1;
          if (s_csum[mid] <= p) lo = mid + 1; else hi = mid;
        }
        int idx = (lo < kT) ? lo : (kT - 1);
        v = *((const vfloat4*)(xrow + (size_t)idx * kD) + lane);
      }
      // streaming (non-temporal) b128 store: output is write-once
      __builtin_nontemporal_store(v, (vfloat4*)(orow + (size_t)p * kD) + lane);
    }
  }
}

// ---------------------------------------------------------------------------
// Host launcher
// ---------------------------------------------------------------------------
extern "C" void kernel_launch(void* const* d_in, const int* in_sizes, int n_in,
                              void* d_out, int out_size, void* d_ws, size_t ws_size,
                              hipStream_t stream) {
  const float* xs = (const float*)d_in[0];
  const int*   ds = (const int*)d_in[1];
  // d_in[2] holds t_out on device; during graph capture we cannot read it,
  // but out_size == B * t_out * D lets us recover it on the host.
  float* out = (float*)d_out;
  int*   csum = (int*)d_ws;  // B*T ints = 64 KB of scratch

  const int t_out = out_size / (kB * kD);
  if (t_out <= 0) return;  // empty output (all-zero durations)

  lenchg_scan_kernel<<<dim3(kB), dim3(256), 0, stream>>>(ds, csum);

  dim3 grid((t_out + FRAMES_PER_BLOCK - 1) / FRAMES_PER_BLOCK, kB);
  lenchg_expand_kernel<<<grid, dim3(256), 0, stream>>>(xs, csum, out, t_out);
}